// KarateGCN_88424786690099
// MI455X (gfx1250) — compile-verified
//
#include <hip/hip_runtime.h>
#include <hip/hip_bf16.h>

typedef float v2f __attribute__((ext_vector_type(2)));
typedef float v8f __attribute__((ext_vector_type(8)));

#define F_IN 128
#define F_HID 128
#define F_OUT 16

static __host__ int cdiv(int a, int b) { return (a + b - 1) / b; }

// ---------------- degree / norm ----------------

__global__ __launch_bounds__(256) void init_deg_kernel(float* deg, int n) {
    int i = blockIdx.x * blockDim.x + threadIdx.x;
    if (i < n) deg[i] = 1.0f;  // self-loop
}

__global__ __launch_bounds__(256) void deg_edges_kernel(const int* __restrict__ dst, float* deg, int e) {
    int i = blockIdx.x * blockDim.x + threadIdx.x;
    if (i < e) unsafeAtomicAdd(&deg[dst[i]], 1.0f);
}

__global__ __launch_bounds__(256) void dinv_kernel(float* deg, int n) {
    int i = blockIdx.x * blockDim.x + threadIdx.x;
    if (i < n) deg[i] = __frsqrt_rn(deg[i]);  // deg >= 1 always (self loop)
}

// ---------------- GEMM1: H[M,128] = X[M,128] @ W[128,128] (f32 WMMA) ----------------
// One wave handles a 16-row tile across all 8 column tiles.

__global__ __launch_bounds__(256) void gemm1_wmma_kernel(const float* __restrict__ X,
                                                         const float* __restrict__ W,
                                                         float* __restrict__ H, int M) {
    int wave = (blockIdx.x * blockDim.x + threadIdx.x) >> 5;
    int lane = threadIdx.x & 31;
    int half = lane >> 4;
    int r    = lane & 15;
    int m0   = wave * 16;
    if (m0 >= M) return;

    v8f acc[8];
#pragma unroll
    for (int i = 0; i < 8; ++i) acc[i] = (v8f)0.0f;

    const float* xrow = X + (size_t)(m0 + r) * F_IN;

    for (int k = 0; k < F_IN; k += 4) {
        int kk = k + half * 2;
        v2f a;
        a.x = xrow[kk];
        a.y = xrow[kk + 1];
#pragma unroll
        for (int nt = 0; nt < 8; ++nt) {
            int col = nt * 16 + r;
            v2f b;
            b.x = W[(size_t)kk * F_HID + col];
            b.y = W[(size_t)(kk + 1) * F_HID + col];
            acc[nt] = __builtin_amdgcn_wmma_f32_16x16x4_f32(
                false, a, false, b, (short)0, acc[nt], false, false);
        }
    }

#pragma unroll
    for (int nt = 0; nt < 8; ++nt) {
#pragma unroll
        for (int rr = 0; rr < 8; ++rr) {
            int row = m0 + rr + half * 8;
            H[(size_t)row * F_HID + nt * 16 + r] = acc[nt][rr];
        }
    }
}

// ---------------- GEMM2: H2[M,16] = A[M,128] @ W2[128,16] (f32 WMMA) ----------------

__global__ __launch_bounds__(256) void gemm2_wmma_kernel(const float* __restrict__ A,
                                                         const float* __restrict__ W2,
                                                         float* __restrict__ H2, int M) {
    int wave = (blockIdx.x * blockDim.x + threadIdx.x) >> 5;
    int lane = threadIdx.x & 31;
    int half = lane >> 4;
    int r    = lane & 15;
    int m0   = wave * 16;
    if (m0 >= M) return;

    v8f acc = (v8f)0.0f;
    const float* arow = A + (size_t)(m0 + r) * F_HID;

    for (int k = 0; k < F_HID; k += 4) {
        int kk = k + half * 2;
        v2f a, b;
        a.x = arow[kk];
        a.y = arow[kk + 1];
        b.x = W2[(size_t)kk * F_OUT + r];
        b.y = W2[(size_t)(kk + 1) * F_OUT + r];
        acc = __builtin_amdgcn_wmma_f32_16x16x4_f32(
            false, a, false, b, (short)0, acc, false, false);
    }

#pragma unroll
    for (int rr = 0; rr < 8; ++rr) {
        int row = m0 + rr + half * 8;
        H2[(size_t)row * F_OUT + r] = acc[rr];
    }
}

// ---------------- aggregation layer 1 (128 features) ----------------

// self-loop contribution also fully initializes out1 (clears poison)
__global__ __launch_bounds__(256) void selfloop1_kernel(const float* __restrict__ h,
                                                        const float* __restrict__ dinv,
                                                        float* __restrict__ out1, int n) {
    int idx = blockIdx.x * blockDim.x + threadIdx.x;
    if (idx >= n * F_HID) return;
    int i = idx >> 7;
    float di = dinv[i];
    out1[idx] = h[idx] * di * di;
}

// one wave per edge: 32 lanes x float4 = 128 features
__global__ __launch_bounds__(256) void edge_agg1_kernel(const int* __restrict__ src,
                                                        const int* __restrict__ dst,
                                                        const float* __restrict__ h,
                                                        const float* __restrict__ dinv,
                                                        float* __restrict__ out1, int e) {
    int wid  = (blockIdx.x * blockDim.x + threadIdx.x) >> 5;
    int lane = threadIdx.x & 31;
    if (wid >= e) return;
    int s = src[wid];
    int d = dst[wid];
    float norm = dinv[s] * dinv[d];
    const float4* hp = (const float4*)(h + (size_t)s * F_HID);
    float4 v = hp[lane];
    float* op = out1 + (size_t)d * F_HID + lane * 4;
    unsafeAtomicAdd(op + 0, v.x * norm);
    unsafeAtomicAdd(op + 1, v.y * norm);
    unsafeAtomicAdd(op + 2, v.z * norm);
    unsafeAtomicAdd(op + 3, v.w * norm);
}

__global__ __launch_bounds__(256) void finalize1_kernel(float* __restrict__ out1,
                                                        const float* __restrict__ b1, int n) {
    int idx = blockIdx.x * blockDim.x + threadIdx.x;
    if (idx >= n * F_HID) return;
    float v = out1[idx] + b1[idx & (F_HID - 1)];
    out1[idx] = v > 0.0f ? v : 0.0f;
}

// ---------------- aggregation layer 2 (16 features) ----------------

__global__ __launch_bounds__(256) void selfloop2_kernel(const float* __restrict__ h2,
                                                        const float* __restrict__ dinv,
                                                        float* __restrict__ out, int n) {
    int idx = blockIdx.x * blockDim.x + threadIdx.x;
    if (idx >= n * F_OUT) return;
    int i = idx >> 4;
    float di = dinv[i];
    out[idx] = h2[idx] * di * di;
}

__global__ __launch_bounds__(256) void edge_agg2_kernel(const int* __restrict__ src,
                                                        const int* __restrict__ dst,
                                                        const float* __restrict__ h2,
                                                        const float* __restrict__ dinv,
                                                        float* __restrict__ out, int e) {
    long long idx = (long long)blockIdx.x * blockDim.x + threadIdx.x;
    if (idx >= (long long)e * F_OUT) return;
    int ei = (int)(idx >> 4);
    int f  = (int)(idx & 15);
    int s = src[ei];
    int d = dst[ei];
    float norm = dinv[s] * dinv[d];
    unsafeAtomicAdd(out + (size_t)d * F_OUT + f, h2[(size_t)s * F_OUT + f] * norm);
}

__global__ __launch_bounds__(256) void finalize2_kernel(float* __restrict__ out,
                                                        const float* __restrict__ b2, int n) {
    int idx = blockIdx.x * blockDim.x + threadIdx.x;
    if (idx >= n * F_OUT) return;
    out[idx] += b2[idx & (F_OUT - 1)];
}

// ---------------- launch ----------------

extern "C" void kernel_launch(void* const* d_in, const int* in_sizes, int n_in,
                              void* d_out, int out_size, void* d_ws, size_t ws_size,
                              hipStream_t stream) {
    const float* x   = (const float*)d_in[0];
    const int*   ei  = (const int*)d_in[1];
    const float* W1  = (const float*)d_in[2];
    const float* b1  = (const float*)d_in[3];
    const float* W2  = (const float*)d_in[4];
    const float* b2  = (const float*)d_in[5];
    float* out = (float*)d_out;

    const int N = in_sizes[0] / F_IN;      // 100000
    const int E = in_sizes[1] / 2;         // 1600000
    const int* src = ei;
    const int* dst = ei + E;

    // workspace layout (bytes), 256-aligned segments
    char* ws = (char*)d_ws;
    size_t off = 0;
    float* dinv = (float*)(ws + off); off += ((size_t)N * 4 + 255) & ~255ULL;
    float* h    = (float*)(ws + off); off += ((size_t)N * F_HID * 4 + 255) & ~255ULL;
    float* out1 = (float*)(ws + off); off += ((size_t)N * F_HID * 4 + 255) & ~255ULL;
    float* h2   = (float*)(ws + off); off += ((size_t)N * F_OUT * 4 + 255) & ~255ULL;
    (void)ws_size;

    const int TB = 256;

    // degrees + symmetric norm
    init_deg_kernel<<<cdiv(N, TB), TB, 0, stream>>>(dinv, N);
    deg_edges_kernel<<<cdiv(E, TB), TB, 0, stream>>>(dst, dinv, E);
    dinv_kernel<<<cdiv(N, TB), TB, 0, stream>>>(dinv, N);

    // layer 1
    {
        int waves = cdiv(N, 16);
        gemm1_wmma_kernel<<<cdiv(waves * 32, TB), TB, 0, stream>>>(x, W1, h, N);
    }
    selfloop1_kernel<<<cdiv(N * F_HID, TB), TB, 0, stream>>>(h, dinv, out1, N);
    edge_agg1_kernel<<<cdiv(E * 32, TB), TB, 0, stream>>>(src, dst, h, dinv, out1, E);
    finalize1_kernel<<<cdiv(N * F_HID, TB), TB, 0, stream>>>(out1, b1, N);

    // layer 2
    {
        int waves = cdiv(N, 16);
        gemm2_wmma_kernel<<<cdiv(waves * 32, TB), TB, 0, stream>>>(out1, W2, h2, N);
    }
    selfloop2_kernel<<<cdiv(N * F_OUT, TB), TB, 0, stream>>>(h2, dinv, out, N);
    {
        long long threads = (long long)E * F_OUT;
        int blocks = (int)((threads + TB - 1) / TB);
        edge_agg2_kernel<<<blocks, TB, 0, stream>>>(src, dst, h2, dinv, out, E);
    }
    finalize2_kernel<<<cdiv(N * F_OUT, TB), TB, 0, stream>>>(out, b2, N);
}